// GCN_3599182594389
// MI455X (gfx1250) — compile-verified
//
#include <hip/hip_runtime.h>
#include <hip/hip_bf16.h>

typedef __attribute__((ext_vector_type(2))) float v2f;
typedef __attribute__((ext_vector_type(8))) float v8f;

#define NEG_SLOPE 0.2f

// ---------------- helpers ----------------

__device__ __forceinline__ float lrelu(float v) {
    return v >= 0.0f ? v : NEG_SLOPE * v;
}

// Order-preserving float -> uint key for atomicMax on floats.
__device__ __forceinline__ unsigned fkey(float f) {
    unsigned u = __float_as_uint(f);
    return (u & 0x80000000u) ? ~u : (u | 0x80000000u);
}
__device__ __forceinline__ float fdekey(unsigned k) {
    return (k & 0x80000000u) ? __uint_as_float(k ^ 0x80000000u)
                             : __uint_as_float(~k);
}

__global__ void zero_u32(unsigned* __restrict__ p, int n) {
    int i = blockIdx.x * blockDim.x + threadIdx.x;
    if (i < n) p[i] = 0u;
}

// ---------------- GEMM1: h1 = x @ W1  [N,128]x[128,256] ----------------
// 1 wave per 16x64 tile: 4 WMMA accumulators share one A fragment.
__global__ __launch_bounds__(32)
void gemm1_wmma(const float* __restrict__ x, const float* __restrict__ W1,
                float* __restrict__ h1, int N) {
    const int mtile = blockIdx.x;
    const int lane  = threadIdx.x;
    const int mrow  = mtile * 16 + (lane & 15);
    const int lrow  = mrow < N ? mrow : N - 1;      // clamp loads, keep EXEC full
    const int khalf = (lane >> 4) * 2;              // 0 or 2
    const int colbase = blockIdx.y * 64;
    const int nc    = lane & 15;

    v8f c0 = {}, c1 = {}, c2 = {}, c3 = {};
    const float* xrow = x + (size_t)lrow * 128;

    for (int k = 0; k < 128; k += 4) {
        const int kb = k + khalf;
        v2f a; a.x = xrow[kb]; a.y = xrow[kb + 1];
        const float* w0 = W1 + (size_t)kb * 256 + colbase + nc;
        const float* w1 = w0 + 256;
        v2f b0; b0.x = w0[0];  b0.y = w1[0];
        v2f b1; b1.x = w0[16]; b1.y = w1[16];
        v2f b2; b2.x = w0[32]; b2.y = w1[32];
        v2f b3; b3.x = w0[48]; b3.y = w1[48];
        c0 = __builtin_amdgcn_wmma_f32_16x16x4_f32(false, a, false, b0, (short)0, c0, false, false);
        c1 = __builtin_amdgcn_wmma_f32_16x16x4_f32(false, a, false, b1, (short)0, c1, false, false);
        c2 = __builtin_amdgcn_wmma_f32_16x16x4_f32(false, a, false, b2, (short)0, c2, false, false);
        c3 = __builtin_amdgcn_wmma_f32_16x16x4_f32(false, a, false, b3, (short)0, c3, false, false);
    }

    const int rowoff = (lane >= 16) ? 8 : 0;
    for (int r = 0; r < 8; ++r) {
        const int grow = mtile * 16 + rowoff + r;
        if (grow < N) {
            float* op = h1 + (size_t)grow * 256 + colbase + nc;
            op[0]  = c0[r];
            op[16] = c1[r];
            op[32] = c2[r];
            op[48] = c3[r];
        }
    }
}

// ---------------- GEMM2: h2 = y1 @ W2  [N,256]x[256,16] ----------------
__global__ __launch_bounds__(32)
void gemm2_wmma(const float* __restrict__ y1, const float* __restrict__ W2,
                float* __restrict__ h2, int N) {
    const int mtile = blockIdx.x;
    const int lane  = threadIdx.x;
    const int mrow  = mtile * 16 + (lane & 15);
    const int lrow  = mrow < N ? mrow : N - 1;
    const int khalf = (lane >> 4) * 2;
    const int nc    = lane & 15;

    v8f c = {};
    const float* yr = y1 + (size_t)lrow * 256;
    for (int k = 0; k < 256; k += 4) {
        const int kb = k + khalf;
        v2f a; a.x = yr[kb]; a.y = yr[kb + 1];
        v2f b; b.x = W2[(size_t)kb * 16 + nc]; b.y = W2[(size_t)(kb + 1) * 16 + nc];
        c = __builtin_amdgcn_wmma_f32_16x16x4_f32(false, a, false, b, (short)0, c, false, false);
    }
    const int rowoff = (lane >= 16) ? 8 : 0;
    for (int r = 0; r < 8; ++r) {
        const int grow = mtile * 16 + rowoff + r;
        if (grow < N) h2[(size_t)grow * 16 + nc] = c[r];
    }
}

// ---------------- attention scores ----------------
// layer1: one thread per (node, head): dot over 64 channels
__global__ void att1_kernel(const float* __restrict__ h1,
                            const float* __restrict__ att_src,
                            const float* __restrict__ att_dst,
                            float* __restrict__ a_src, float* __restrict__ a_dst,
                            int NH) {
    int i = blockIdx.x * blockDim.x + threadIdx.x;
    if (i >= NH) return;
    const int h = i & 3;
    const float* hp = h1 + (size_t)(i >> 2) * 256 + h * 64;
    const float* as = att_src + h * 64;
    const float* ad = att_dst + h * 64;
    float s = 0.0f, d = 0.0f;
    for (int c = 0; c < 64; ++c) {
        const float v = hp[c];
        s += v * as[c];
        d += v * ad[c];
    }
    a_src[i] = s;
    a_dst[i] = d;
}

// layer2: one thread per node, H=1, C=16
__global__ void att2_kernel(const float* __restrict__ h2,
                            const float* __restrict__ att_src,
                            const float* __restrict__ att_dst,
                            float* __restrict__ a_src, float* __restrict__ a_dst,
                            int N) {
    int i = blockIdx.x * blockDim.x + threadIdx.x;
    if (i >= N) return;
    const float* hp = h2 + (size_t)i * 16;
    float s = 0.0f, d = 0.0f;
    for (int c = 0; c < 16; ++c) {
        const float v = hp[c];
        s += v * att_src[c];
        d += v * att_dst[c];
    }
    a_src[i] = s;
    a_dst[i] = d;
}

// ---------------- edge passes, layer1 (H=4) ----------------
__device__ __forceinline__ void edge_sd(const int* __restrict__ ei, int e, int E,
                                        int& s, int& d) {
    if (e < E) { s = ei[e]; d = ei[E + e]; }
    else       { s = d = e - E; }           // self-loops appended
}

__global__ void edge_max1(const int* __restrict__ ei, int E, int N,
                          const float* __restrict__ a_src, const float* __restrict__ a_dst,
                          unsigned* __restrict__ mkey) {
    int e = blockIdx.x * blockDim.x + threadIdx.x;
    const int ET = E + N;
    if (e >= ET) return;
    int s, d; edge_sd(ei, e, E, s, d);
    for (int h = 0; h < 4; ++h) {
        const float v = lrelu(a_src[s * 4 + h] + a_dst[d * 4 + h]);
        atomicMax(&mkey[d * 4 + h], fkey(v));
    }
}

__global__ void edge_expsum1(const int* __restrict__ ei, int E, int N,
                             const float* __restrict__ a_src, const float* __restrict__ a_dst,
                             const unsigned* __restrict__ mkey,
                             float* __restrict__ ex, float* __restrict__ den) {
    int e = blockIdx.x * blockDim.x + threadIdx.x;
    const int ET = E + N;
    if (e >= ET) return;
    int s, d; edge_sd(ei, e, E, s, d);
    for (int h = 0; h < 4; ++h) {
        const float v  = lrelu(a_src[s * 4 + h] + a_dst[d * 4 + h]);
        const float m  = fdekey(mkey[d * 4 + h]);
        const float xv = expf(v - m);
        ex[(size_t)e * 4 + h] = xv;
        atomicAdd(&den[d * 4 + h], xv);
    }
}

// 64 threads per edge, each handles 4 channels (float4 gather, 4 f32 atomics)
__global__ void edge_agg1(const int* __restrict__ ei, int E, int N,
                          const float* __restrict__ ex, const float* __restrict__ den,
                          const float* __restrict__ h1, float* __restrict__ out1) {
    const int gid = blockIdx.x * blockDim.x + threadIdx.x;
    const int e = gid >> 6;
    const int t = gid & 63;
    const int ET = E + N;
    if (e >= ET) return;
    int s, d; edge_sd(ei, e, E, s, d);
    const int head = t >> 4;                     // 64 channels per head / 4 per thread
    const float alpha = ex[(size_t)e * 4 + head] / den[d * 4 + head];
    const float4 hv = *(const float4*)(h1 + (size_t)s * 256 + t * 4);
    float* op = out1 + (size_t)d * 256 + t * 4;
    atomicAdd(op + 0, alpha * hv.x);
    atomicAdd(op + 1, alpha * hv.y);
    atomicAdd(op + 2, alpha * hv.z);
    atomicAdd(op + 3, alpha * hv.w);
}

// in-place: y = relu(agg + b1)
__global__ void relu_bias1(float* __restrict__ out1, const float* __restrict__ b1, int n) {
    int i = blockIdx.x * blockDim.x + threadIdx.x;
    if (i < n) out1[i] = fmaxf(out1[i] + b1[i & 255], 0.0f);
}

// ---------------- edge passes, layer2 (H=1, C=16) ----------------
__global__ void edge_max2(const int* __restrict__ ei, int E, int N,
                          const float* __restrict__ a_src, const float* __restrict__ a_dst,
                          unsigned* __restrict__ mkey) {
    int e = blockIdx.x * blockDim.x + threadIdx.x;
    const int ET = E + N;
    if (e >= ET) return;
    int s, d; edge_sd(ei, e, E, s, d);
    const float v = lrelu(a_src[s] + a_dst[d]);
    atomicMax(&mkey[d], fkey(v));
}

__global__ void edge_expsum2(const int* __restrict__ ei, int E, int N,
                             const float* __restrict__ a_src, const float* __restrict__ a_dst,
                             const unsigned* __restrict__ mkey,
                             float* __restrict__ ex, float* __restrict__ den) {
    int e = blockIdx.x * blockDim.x + threadIdx.x;
    const int ET = E + N;
    if (e >= ET) return;
    int s, d; edge_sd(ei, e, E, s, d);
    const float v  = lrelu(a_src[s] + a_dst[d]);
    const float xv = expf(v - fdekey(mkey[d]));
    ex[e] = xv;
    atomicAdd(&den[d], xv);
}

// 16 threads per edge, one channel each
__global__ void edge_agg2(const int* __restrict__ ei, int E, int N,
                          const float* __restrict__ ex, const float* __restrict__ den,
                          const float* __restrict__ h2, float* __restrict__ out) {
    const int gid = blockIdx.x * blockDim.x + threadIdx.x;
    const int e = gid >> 4;
    const int t = gid & 15;
    const int ET = E + N;
    if (e >= ET) return;
    int s, d; edge_sd(ei, e, E, s, d);
    const float alpha = ex[e] / den[d];
    atomicAdd(&out[(size_t)d * 16 + t], alpha * h2[(size_t)s * 16 + t]);
}

__global__ void bias2_kernel(float* __restrict__ out, const float* __restrict__ b2, int n) {
    int i = blockIdx.x * blockDim.x + threadIdx.x;
    if (i < n) out[i] += b2[i & 15];
}

// ---------------- launch ----------------

extern "C" void kernel_launch(void* const* d_in, const int* in_sizes, int n_in,
                              void* d_out, int out_size, void* d_ws, size_t ws_size,
                              hipStream_t stream) {
    const float* x   = (const float*)d_in[0];
    const int*   ei  = (const int*)  d_in[1];
    const float* W1  = (const float*)d_in[2];
    const float* as1 = (const float*)d_in[3];
    const float* ad1 = (const float*)d_in[4];
    const float* b1  = (const float*)d_in[5];
    const float* W2  = (const float*)d_in[6];
    const float* as2 = (const float*)d_in[7];
    const float* ad2 = (const float*)d_in[8];
    const float* b2  = (const float*)d_in[9];

    const int N  = in_sizes[0] / 128;   // 50000
    const int E  = in_sizes[1] / 2;     // 600000
    const int ET = E + N;               // edges + self loops

    // carve workspace (256B aligned chunks)
    char* w = (char*)d_ws;
    auto carve = [&](size_t bytes) -> void* {
        void* p = (void*)w;
        w += (bytes + 255) & ~((size_t)255);
        return p;
    };
    float*    h1    = (float*)   carve((size_t)N * 256 * 4);
    float*    out1  = (float*)   carve((size_t)N * 256 * 4);
    float*    aS1   = (float*)   carve((size_t)N * 4 * 4);
    float*    aD1   = (float*)   carve((size_t)N * 4 * 4);
    unsigned* mkey1 = (unsigned*)carve((size_t)N * 4 * 4);
    float*    den1  = (float*)   carve((size_t)N * 4 * 4);
    float*    ex1   = (float*)   carve((size_t)ET * 4 * 4);
    float*    h2    = (float*)   carve((size_t)N * 16 * 4);
    float*    aS2   = (float*)   carve((size_t)N * 4);
    float*    aD2   = (float*)   carve((size_t)N * 4);
    unsigned* mkey2 = (unsigned*)carve((size_t)N * 4);
    float*    den2  = (float*)   carve((size_t)N * 4);
    float*    ex2   = (float*)   carve((size_t)ET * 4);

    const int TB = 256;
    auto blocks = [&](long long n) { return (unsigned)((n + TB - 1) / TB); };

    // zero-init accumulators / softmax state / output
    zero_u32<<<blocks((long long)N * 256), TB, 0, stream>>>((unsigned*)out1, N * 256);
    zero_u32<<<blocks(N * 4), TB, 0, stream>>>(mkey1, N * 4);
    zero_u32<<<blocks(N * 4), TB, 0, stream>>>((unsigned*)den1, N * 4);
    zero_u32<<<blocks(N), TB, 0, stream>>>(mkey2, N);
    zero_u32<<<blocks(N), TB, 0, stream>>>((unsigned*)den2, N);
    zero_u32<<<blocks(N * 16), TB, 0, stream>>>((unsigned*)d_out, N * 16);

    // layer 1
    dim3 g1((N + 15) / 16, 4);
    gemm1_wmma<<<g1, 32, 0, stream>>>(x, W1, h1, N);
    att1_kernel<<<blocks(N * 4), TB, 0, stream>>>(h1, as1, ad1, aS1, aD1, N * 4);
    edge_max1<<<blocks(ET), TB, 0, stream>>>(ei, E, N, aS1, aD1, mkey1);
    edge_expsum1<<<blocks(ET), TB, 0, stream>>>(ei, E, N, aS1, aD1, mkey1, ex1, den1);
    edge_agg1<<<blocks((long long)ET * 64), TB, 0, stream>>>(ei, E, N, ex1, den1, h1, out1);
    relu_bias1<<<blocks((long long)N * 256), TB, 0, stream>>>(out1, b1, N * 256);

    // layer 2
    gemm2_wmma<<<(N + 15) / 16, 32, 0, stream>>>(out1, W2, h2, N);
    att2_kernel<<<blocks(N), TB, 0, stream>>>(h2, as2, ad2, aS2, aD2, N);
    edge_max2<<<blocks(ET), TB, 0, stream>>>(ei, E, N, aS2, aD2, mkey2);
    edge_expsum2<<<blocks(ET), TB, 0, stream>>>(ei, E, N, aS2, aD2, mkey2, ex2, den2);
    edge_agg2<<<blocks((long long)ET * 16), TB, 0, stream>>>(ei, E, N, ex2, den2, h2, (float*)d_out);
    bias2_kernel<<<blocks(N * 16), TB, 0, stream>>>((float*)d_out, b2, N * 16);
}